// GNNModel_34462817583810
// MI455X (gfx1250) — compile-verified
//
#include <hip/hip_runtime.h>

#define NNODES 100000
#define NEDGES 1600000
#define FIN 128
#define FH 64
#define FOUT 32

typedef float v2f __attribute__((ext_vector_type(2)));
typedef float v8f __attribute__((ext_vector_type(8)));

static __device__ __forceinline__ v8f wmma_f32_k4(v2f a, v2f b, v8f c) {
  // V_WMMA_F32_16X16X4_F32: D = A(16x4) * B(4x16) + C(16x16), full fp32
  return __builtin_amdgcn_wmma_f32_16x16x4_f32(false, a, false, b, (short)0, c,
                                               false, false);
}

// ---------------------------------------------------------------- degree/norm
__global__ void deg_init_kernel(float* __restrict__ deg) {
  int i = blockIdx.x * blockDim.x + threadIdx.x;
  if (i < NNODES) deg[i] = 1.0f;  // self-loop
}

__global__ void deg_count_kernel(const int* __restrict__ dst,
                                 float* __restrict__ deg) {
  int i = blockIdx.x * blockDim.x + threadIdx.x;
  if (i < NEDGES) unsafeAtomicAdd(&deg[dst[i]], 1.0f);
}

__global__ void dinv_kernel(const float* __restrict__ deg,
                            float* __restrict__ dinv) {
  int i = blockIdx.x * blockDim.x + threadIdx.x;
  if (i < NNODES) dinv[i] = rsqrtf(deg[i]);  // deg >= 1 always (self-loop)
}

// ---------------------------------------------------------------- GEMM layer 1
// h1[N,64] = x[N,128] @ W1[64,128]^T    (fp32 WMMA, W1 staged in LDS)
__global__ __launch_bounds__(256) void gcn_gemm1(const float* __restrict__ x,
                                                 const float* __restrict__ W1,
                                                 float* __restrict__ h1) {
  __shared__ float sW[FH * FIN];  // 32 KB
  for (int i = threadIdx.x; i < FH * FIN; i += 256) sW[i] = W1[i];
  __syncthreads();

  const int lane = threadIdx.x & 31;
  const int wave = threadIdx.x >> 5;
  const int tile = blockIdx.x * 8 + wave;       // 16-row M tile (wave-uniform)
  if (tile * 16 >= NNODES) return;              // whole wave exits -> EXEC ok
  const int m0 = tile * 16;
  const int lrow = lane & 15;                   // M (for A) / N (for B) index
  const int koff = (lane >> 4) << 1;            // lanes 0-15: K={0,1}; 16-31: K={2,3}

  v8f acc0 = {}, acc1 = {}, acc2 = {}, acc3 = {};
  const float* xrow = x + (size_t)(m0 + lrow) * FIN;
#pragma unroll 4
  for (int k = 0; k < FIN; k += 4) {
    v2f a = *(const v2f*)(xrow + k + koff);  // A[16x4] fragment
    // B[4x16] fragments: B[k][n] = W1[n][k] -> contiguous float2 in W1 row
    v2f b0 = *(const v2f*)(&sW[(0 * 16 + lrow) * FIN + k + koff]);
    v2f b1 = *(const v2f*)(&sW[(1 * 16 + lrow) * FIN + k + koff]);
    v2f b2 = *(const v2f*)(&sW[(2 * 16 + lrow) * FIN + k + koff]);
    v2f b3 = *(const v2f*)(&sW[(3 * 16 + lrow) * FIN + k + koff]);
    acc0 = wmma_f32_k4(a, b0, acc0);
    acc1 = wmma_f32_k4(a, b1, acc1);
    acc2 = wmma_f32_k4(a, b2, acc2);
    acc3 = wmma_f32_k4(a, b3, acc3);
  }
  // D layout: VGPR r -> row m0+r (lanes 0-15) / m0+8+r (lanes 16-31), col = lane%16
  const int rbase = m0 + ((lane >> 4) << 3);
#pragma unroll
  for (int r = 0; r < 8; ++r) {
    h1[(size_t)(rbase + r) * FH + 0 * 16 + lrow] = acc0[r];
    h1[(size_t)(rbase + r) * FH + 1 * 16 + lrow] = acc1[r];
    h1[(size_t)(rbase + r) * FH + 2 * 16 + lrow] = acc2[r];
    h1[(size_t)(rbase + r) * FH + 3 * 16 + lrow] = acc3[r];
  }
}

// ---------------------------------------------------------------- GEMM layer 2
// h2[N,32] = relu(agg1 + b1) @ W2[32,64]^T   (ReLU+bias fused into A load)
__global__ __launch_bounds__(256) void gcn_gemm2(const float* __restrict__ agg1,
                                                 const float* __restrict__ b1,
                                                 const float* __restrict__ W2,
                                                 float* __restrict__ h2) {
  __shared__ float sW[FOUT * FH];  // 8 KB
  __shared__ float sB[FH];
  for (int i = threadIdx.x; i < FOUT * FH; i += 256) sW[i] = W2[i];
  if (threadIdx.x < FH) sB[threadIdx.x] = b1[threadIdx.x];
  __syncthreads();

  const int lane = threadIdx.x & 31;
  const int wave = threadIdx.x >> 5;
  const int tile = blockIdx.x * 8 + wave;
  if (tile * 16 >= NNODES) return;
  const int m0 = tile * 16;
  const int lrow = lane & 15;
  const int koff = (lane >> 4) << 1;

  v8f acc0 = {}, acc1 = {};
  const float* arow = agg1 + (size_t)(m0 + lrow) * FH;
#pragma unroll 4
  for (int k = 0; k < FH; k += 4) {
    v2f ar = *(const v2f*)(arow + k + koff);
    v2f a;
    a.x = fmaxf(ar.x + sB[k + koff + 0], 0.0f);
    a.y = fmaxf(ar.y + sB[k + koff + 1], 0.0f);
    v2f b0 = *(const v2f*)(&sW[(0 * 16 + lrow) * FH + k + koff]);
    v2f b1v = *(const v2f*)(&sW[(1 * 16 + lrow) * FH + k + koff]);
    acc0 = wmma_f32_k4(a, b0, acc0);
    acc1 = wmma_f32_k4(a, b1v, acc1);
  }
  const int rbase = m0 + ((lane >> 4) << 3);
#pragma unroll
  for (int r = 0; r < 8; ++r) {
    h2[(size_t)(rbase + r) * FOUT + 0 * 16 + lrow] = acc0[r];
    h2[(size_t)(rbase + r) * FOUT + 1 * 16 + lrow] = acc1[r];
  }
}

// ---------------------------------------------------------------- aggregation
// Self-loop contribution written directly (no atomics): agg[i] = h[i]*dinv[i]^2
__global__ void agg1_init_kernel(const float* __restrict__ h1,
                                 const float* __restrict__ dinv,
                                 float* __restrict__ agg1) {
  int i = blockIdx.x * blockDim.x + threadIdx.x;  // over N*FH/4 float4s
  if (i >= NNODES * (FH / 4)) return;
  int node = i / (FH / 4);
  float w = dinv[node];
  w *= w;
  float4 v = ((const float4*)h1)[i];
  float4 o = {v.x * w, v.y * w, v.z * w, v.w * w};
  ((float4*)agg1)[i] = o;
}

// Self-loop + bias baked into the output buffer for layer 2
__global__ void agg2_init_kernel(const float* __restrict__ h2,
                                 const float* __restrict__ dinv,
                                 const float* __restrict__ b2,
                                 float* __restrict__ out) {
  int i = blockIdx.x * blockDim.x + threadIdx.x;  // over N*FOUT/4 float4s
  if (i >= NNODES * (FOUT / 4)) return;
  int node = i / (FOUT / 4);
  int c4 = i % (FOUT / 4);
  float w = dinv[node];
  w *= w;
  float4 v = ((const float4*)h2)[i];
  float4 bb = ((const float4*)b2)[c4];
  float4 o = {v.x * w + bb.x, v.y * w + bb.y, v.z * w + bb.z, v.w * w + bb.w};
  ((float4*)out)[i] = o;
}

// Per-edge gather/scale/scatter-add: F/4 threads per edge, float4 gather,
// 4 hardware fp32 atomics (L2-resident for this problem size).
template <int F>
__global__ void agg_edges_kernel(const int* __restrict__ src,
                                 const int* __restrict__ dst,
                                 const float* __restrict__ dinv,
                                 const float* __restrict__ h,
                                 float* __restrict__ agg) {
  const int TPE = F / 4;
  long long t = (long long)blockIdx.x * blockDim.x + threadIdx.x;
  long long e = t / TPE;
  int c = (int)(t % TPE);
  if (e >= NEDGES) return;
  int s = src[e];
  int d = dst[e];
  float w = dinv[s] * dinv[d];
  float4 v = ((const float4*)(h + (size_t)s * F))[c];
  float* o = agg + (size_t)d * F + c * 4;
  unsafeAtomicAdd(o + 0, v.x * w);
  unsafeAtomicAdd(o + 1, v.y * w);
  unsafeAtomicAdd(o + 2, v.z * w);
  unsafeAtomicAdd(o + 3, v.w * w);
}

// ---------------------------------------------------------------- launch
extern "C" void kernel_launch(void* const* d_in, const int* in_sizes, int n_in,
                              void* d_out, int out_size, void* d_ws,
                              size_t ws_size, hipStream_t stream) {
  const float* x = (const float*)d_in[0];
  const int* edge_index = (const int*)d_in[1];
  const float* W1 = (const float*)d_in[2];
  const float* b1 = (const float*)d_in[3];
  const float* W2 = (const float*)d_in[4];
  const float* b2 = (const float*)d_in[5];
  float* out = (float*)d_out;

  const int* src = edge_index;            // edge_index[0, :]
  const int* dst = edge_index + NEDGES;   // edge_index[1, :]

  // Workspace layout (floats): deg | dinv | h1 | agg1 | h2  (~65 MB total)
  float* ws = (float*)d_ws;
  float* deg = ws;
  float* dinv = ws + NNODES;
  float* h1 = ws + 2 * (size_t)NNODES;
  float* agg1 = h1 + (size_t)NNODES * FH;
  float* h2 = agg1 + (size_t)NNODES * FH;

  const int B = 256;
  const int nTiles = NNODES / 16;                 // 6250, exact
  const int gemmBlocks = (nTiles + 7) / 8;        // 8 waves/block

  deg_init_kernel<<<(NNODES + B - 1) / B, B, 0, stream>>>(deg);
  deg_count_kernel<<<(NEDGES + B - 1) / B, B, 0, stream>>>(dst, deg);
  dinv_kernel<<<(NNODES + B - 1) / B, B, 0, stream>>>(deg, dinv);

  gcn_gemm1<<<gemmBlocks, B, 0, stream>>>(x, W1, h1);

  agg1_init_kernel<<<(NNODES * (FH / 4) + B - 1) / B, B, 0, stream>>>(h1, dinv,
                                                                      agg1);
  {
    long long tot = (long long)NEDGES * (FH / 4);
    agg_edges_kernel<FH><<<(unsigned)((tot + B - 1) / B), B, 0, stream>>>(
        src, dst, dinv, h1, agg1);
  }

  gcn_gemm2<<<gemmBlocks, B, 0, stream>>>(agg1, b1, W2, h2);

  agg2_init_kernel<<<(NNODES * (FOUT / 4) + B - 1) / B, B, 0, stream>>>(
      h2, dinv, b2, out);
  {
    long long tot = (long long)NEDGES * (FOUT / 4);
    agg_edges_kernel<FOUT><<<(unsigned)((tot + B - 1) / B), B, 0, stream>>>(
        src, dst, dinv, h2, out);
  }
}